// GraphModel_25735444037705
// MI455X (gfx1250) — compile-verified
//
#include <hip/hip_runtime.h>
#include <math.h>

typedef float v2f __attribute__((ext_vector_type(2)));
typedef float v8f __attribute__((ext_vector_type(8)));

#define TPB 256

// ---------------- elementwise helpers ----------------

__global__ void k_fill(float* __restrict__ p, float v, long long n) {
  long long i = (long long)blockIdx.x * blockDim.x + threadIdx.x;
  if (i < n) p[i] = v;
}

__global__ void k_deg_count(const int* __restrict__ dst, long long n,
                            float* __restrict__ deg) {
  long long i = (long long)blockIdx.x * blockDim.x + threadIdx.x;
  if (i < n) atomicAdd(&deg[dst[i]], 1.0f);
}

__global__ void k_rsqrt(float* __restrict__ d, long long n) {
  long long i = (long long)blockIdx.x * blockDim.x + threadIdx.x;
  if (i < n) d[i] = rsqrtf(d[i]);   // deg >= 1 always (self loops)
}

// ---------------- WMMA f32 GEMM: C[M,N] = A[M,K] @ B[K,N] ----------------
// One wave computes one 16x16 tile of C using V_WMMA_F32_16X16X4_F32.
// A-frag (16x4 f32): lanes 0-15 -> rows M=lane, regs = K {k,k+1};
//                    lanes 16-31 -> rows M=lane-16, regs = K {k+2,k+3}.
// B-frag (4x16 f32): lanes 0-15 -> col N=lane, regs = K {k,k+1};
//                    lanes 16-31 -> col N=lane-16, regs = K {k+2,k+3}.
// C/D (16x16 f32):   reg v, lane l -> row m0+v+8*(l>>4), col n0+(l&15).
__global__ void k_gemm_wmma(const float* __restrict__ A, const float* __restrict__ B,
                            float* __restrict__ C, int M, int N, int K) {
  int gwave = (int)(((long long)blockIdx.x * blockDim.x + threadIdx.x) >> 5);
  int lane  = threadIdx.x & 31;
  int ntiles = N >> 4;
  int mtiles = (M + 15) >> 4;
  if (gwave >= ntiles * mtiles) return;        // wave-uniform: EXEC stays all-1s
  int tm = gwave / ntiles, tn = gwave % ntiles;
  int m0 = tm << 4, n0 = tn << 4;
  int lh = lane >> 4;        // 0 or 1 (K-half)
  int lm = lane & 15;
  int arow = m0 + lm;
  arow = arow < M ? arow : (M - 1);            // branchless clamp (no EXEC change)
  const float* Ar = A + (size_t)arow * K;
  v8f acc = {0.f, 0.f, 0.f, 0.f, 0.f, 0.f, 0.f, 0.f};
  for (int k = 0; k < K; k += 4) {
    int ka = k + (lh << 1);
    v2f av, bv;
    av.x = Ar[ka];
    av.y = Ar[ka + 1];
    bv.x = B[(size_t)ka * N + n0 + lm];
    bv.y = B[(size_t)(ka + 1) * N + n0 + lm];
    acc = __builtin_amdgcn_wmma_f32_16x16x4_f32(
        /*neg_a=*/false, av, /*neg_b=*/false, bv,
        /*c_mod=*/(short)0, acc, /*reuse_a=*/false, /*reuse_b=*/false);
  }
#pragma unroll
  for (int v = 0; v < 8; ++v) {
    int row = m0 + v + (lh << 3);
    if (row < M) C[(size_t)row * N + n0 + lm] = acc[v];
  }
}

// ---------------- GCN scatter: out[dst] += h[src] * dinv[src]*dinv[dst] -------
// one thread per (edge, 4-float chunk); F = 1<<fshift floats per row,
// chunks per edge = F/4 = 1<<cshift.
__global__ void k_scatter(const float* __restrict__ h, const int* __restrict__ src,
                          const int* __restrict__ dst, const float* __restrict__ dinv,
                          long long nWork, int cshift, int F, float* __restrict__ out) {
  long long idx = (long long)blockIdx.x * blockDim.x + threadIdx.x;
  if (idx >= nWork) return;
  long long e = idx >> cshift;
  int c = (int)(idx & ((1 << cshift) - 1)) << 2;
  int s = src[e], d = dst[e];
  float nrm = dinv[s] * dinv[d];
  float4 v = *(const float4*)(h + (size_t)s * F + c);
  float* op = out + (size_t)d * F + c;
  atomicAdd(op + 0, v.x * nrm);
  atomicAdd(op + 1, v.y * nrm);
  atomicAdd(op + 2, v.z * nrm);
  atomicAdd(op + 3, v.w * nrm);
}

// agg[i,c] += h[i,c]*dinv[i]^2 + bias[c]; optional sigmoid. total = n*F.
__global__ void k_finalize(const float* __restrict__ h, const float* __restrict__ dinv,
                           const float* __restrict__ bias, float* __restrict__ agg,
                           long long total, int fshift, int act) {
  long long idx = (long long)blockIdx.x * blockDim.x + threadIdx.x;
  if (idx >= total) return;
  long long i = idx >> fshift;
  int c = (int)(idx & ((1 << fshift) - 1));
  float di = dinv[i];
  float v = agg[idx] + h[idx] * di * di + bias[c];
  if (act) v = 1.0f / (1.0f + expf(-v));
  agg[idx] = v;
}

// ---------------- pooling ----------------

__global__ void k_pool_feat(const float* __restrict__ f, const int* __restrict__ seg,
                            long long total, int fshift, float* __restrict__ sums) {
  long long idx = (long long)blockIdx.x * blockDim.x + threadIdx.x;
  if (idx >= total) return;
  long long i = idx >> fshift;
  int c = (int)(idx & ((1 << fshift) - 1));
  atomicAdd(&sums[((size_t)seg[i] << fshift) + c], f[idx]);
}

__global__ void k_pool_feat_edge(const float* __restrict__ f, const int* __restrict__ esrc,
                                 const int* __restrict__ batch, long long total,
                                 int fshift, float* __restrict__ sums) {
  long long idx = (long long)blockIdx.x * blockDim.x + threadIdx.x;
  if (idx >= total) return;
  long long i = idx >> fshift;
  int c = (int)(idx & ((1 << fshift) - 1));
  int b = batch[esrc[i]];
  atomicAdd(&sums[((size_t)b << fshift) + c], f[idx]);
}

__global__ void k_pool_cnt(const int* __restrict__ seg, long long n, float* __restrict__ cnt) {
  long long i = (long long)blockIdx.x * blockDim.x + threadIdx.x;
  if (i < n) atomicAdd(&cnt[seg[i]], 1.0f);
}

__global__ void k_pool_cnt_edge(const int* __restrict__ esrc, const int* __restrict__ batch,
                                long long n, float* __restrict__ cnt) {
  long long i = (long long)blockIdx.x * blockDim.x + threadIdx.x;
  if (i < n) atomicAdd(&cnt[batch[esrc[i]]], 1.0f);
}

// g[b, 0:64] = sumsN/max(cntN,1); g[b, 64:128] = sumsE/max(cntE,1)
__global__ void k_build_g(const float* __restrict__ sN, const float* __restrict__ cN,
                          const float* __restrict__ sE, const float* __restrict__ cE,
                          float* __restrict__ g, int total) {
  int idx = blockIdx.x * blockDim.x + threadIdx.x;
  if (idx >= total) return;
  int b = idx >> 7, c = idx & 127;
  float v = (c < 64) ? sN[b * 64 + c]        / fmaxf(cN[b], 1.0f)
                     : sE[b * 64 + (c - 64)] / fmaxf(cE[b], 1.0f);
  g[idx] = v;
}

__global__ void k_bias_out(const float* __restrict__ t, const float* __restrict__ bfc,
                           float* __restrict__ out, int total) {
  int idx = blockIdx.x * blockDim.x + threadIdx.x;
  if (idx >= total) return;
  out[idx] = t[idx] + bfc[idx & 63];
}

// ---------------- launch ----------------

extern "C" void kernel_launch(void* const* d_in, const int* in_sizes, int n_in,
                              void* d_out, int out_size, void* d_ws, size_t ws_size,
                              hipStream_t stream) {
  const float* x         = (const float*)d_in[0];
  const float* edge_attr = (const float*)d_in[1];
  const int*   ei        = (const int*)d_in[2];
  const int*   eil       = (const int*)d_in[3];
  const int*   batch     = (const int*)d_in[4];
  const float* Wn1 = (const float*)d_in[5];  const float* bn1 = (const float*)d_in[6];
  const float* Wn2 = (const float*)d_in[7];  const float* bn2 = (const float*)d_in[8];
  const float* We1 = (const float*)d_in[9];  const float* be1 = (const float*)d_in[10];
  const float* We2 = (const float*)d_in[11]; const float* be2 = (const float*)d_in[12];
  const float* Wfc = (const float*)d_in[13]; const float* bfc = (const float*)d_in[14];

  const int DINN = 128, DHN = 128, DON = 64, DINE = 64, DHE = 64, DOE = 64;
  long long Nn = in_sizes[0] / DINN;   // 20000
  long long Ee = in_sizes[1] / DINE;   // 100000
  long long Ll = in_sizes[3] / 2;      // line-graph edge count
  int Bb = out_size / DON;             // 64

  const int* srcN = ei;   const int* dstN = ei + Ee;
  const int* srcL = eil;  const int* dstL = eil + Ll;

  // workspace layout (floats)
  float* ws = (float*)d_ws;
  size_t oA = 0;
  size_t oB = oA + (size_t)Nn * 128;
  size_t oC = oB + (size_t)Nn * 128;
  size_t oD = oC + (size_t)Ee * 64;
  size_t oDegN = oD + (size_t)Ee * 64;
  size_t oDegE = oDegN + (size_t)Nn;
  size_t oSumsN = oDegE + (size_t)Ee;
  size_t oCntN  = oSumsN + (size_t)Bb * 64;
  size_t oSumsE = oCntN + (size_t)Bb;
  size_t oCntE  = oSumsE + (size_t)Bb * 64;
  size_t oG     = oCntE + (size_t)Bb;
  size_t oTmp   = oG + (size_t)Bb * 128;

  float* bufA = ws + oA;   float* bufB = ws + oB;
  float* bufC = ws + oC;   float* bufD = ws + oD;
  float* degN = ws + oDegN; float* degE = ws + oDegE;
  float* sumsN = ws + oSumsN; float* cntN = ws + oCntN;
  float* sumsE = ws + oSumsE; float* cntE = ws + oCntE;
  float* g = ws + oG; float* tmp = ws + oTmp;

  auto nblk = [](long long n) { return (unsigned)((n + TPB - 1) / TPB); };
  auto gemm = [&](const float* A, const float* Bm, float* C, long long M, int N, int K) {
    long long waves = (long long)((M + 15) >> 4) * (N >> 4);
    unsigned blk = (unsigned)((waves * 32 + 127) / 128);
    k_gemm_wmma<<<blk, 128, 0, stream>>>(A, Bm, C, (int)M, N, K);
  };

  // ---- degrees: deg = 1 (self loop) + in-degree; dinv = rsqrt(deg) ----
  k_fill<<<nblk(Nn), TPB, 0, stream>>>(degN, 1.0f, Nn);
  k_fill<<<nblk(Ee), TPB, 0, stream>>>(degE, 1.0f, Ee);
  if (Ee) k_deg_count<<<nblk(Ee), TPB, 0, stream>>>(dstN, Ee, degN);
  if (Ll) k_deg_count<<<nblk(Ll), TPB, 0, stream>>>(dstL, Ll, degE);
  k_rsqrt<<<nblk(Nn), TPB, 0, stream>>>(degN, Nn);
  k_rsqrt<<<nblk(Ee), TPB, 0, stream>>>(degE, Ee);

  // ---- node GCN layer 1: h1 = x@Wn1; agg; nf1 = sigmoid(agg + self + bn1) ----
  gemm(x, Wn1, bufA, Nn, DHN, DINN);                       // h1: N x 128
  k_fill<<<nblk(Nn * 128), TPB, 0, stream>>>(bufB, 0.0f, Nn * 128);
  if (Ee) k_scatter<<<nblk(Ee * 32), TPB, 0, stream>>>(bufA, srcN, dstN, degN,
                                                       Ee * 32, 5, 128, bufB);
  k_finalize<<<nblk(Nn * 128), TPB, 0, stream>>>(bufA, degN, bn1, bufB, Nn * 128, 7, 1);

  // ---- node GCN layer 2: h2 = nf1@Wn2; agg; nf2 = agg + self + bn2 ----
  gemm(bufB, Wn2, bufA, Nn, DON, DHN);                     // h2: N x 64
  k_fill<<<nblk(Nn * 64), TPB, 0, stream>>>(bufB, 0.0f, Nn * 64);
  if (Ee) k_scatter<<<nblk(Ee * 16), TPB, 0, stream>>>(bufA, srcN, dstN, degN,
                                                       Ee * 16, 4, 64, bufB);
  k_finalize<<<nblk(Nn * 64), TPB, 0, stream>>>(bufA, degN, bn2, bufB, Nn * 64, 6, 0);
  // nf2 lives in bufB (N x 64)

  // ---- edge GCN layer 1 on line graph ----
  gemm(edge_attr, We1, bufC, Ee, DHE, DINE);               // eh1: E x 64
  k_fill<<<nblk(Ee * 64), TPB, 0, stream>>>(bufD, 0.0f, Ee * 64);
  if (Ll) k_scatter<<<nblk(Ll * 16), TPB, 0, stream>>>(bufC, srcL, dstL, degE,
                                                       Ll * 16, 4, 64, bufD);
  k_finalize<<<nblk(Ee * 64), TPB, 0, stream>>>(bufC, degE, be1, bufD, Ee * 64, 6, 1);

  // ---- edge GCN layer 2 ----
  gemm(bufD, We2, bufC, Ee, DOE, DHE);                     // eh2: E x 64
  k_fill<<<nblk(Ee * 64), TPB, 0, stream>>>(bufD, 0.0f, Ee * 64);
  if (Ll) k_scatter<<<nblk(Ll * 16), TPB, 0, stream>>>(bufC, srcL, dstL, degE,
                                                       Ll * 16, 4, 64, bufD);
  k_finalize<<<nblk(Ee * 64), TPB, 0, stream>>>(bufC, degE, be2, bufD, Ee * 64, 6, 0);
  // ef2 lives in bufD (E x 64)

  // ---- pooling ----
  long long poolLen = (long long)2 * Bb * 64 + 2 * Bb;     // sumsN,cntN,sumsE,cntE contiguous
  k_fill<<<nblk(poolLen), TPB, 0, stream>>>(sumsN, 0.0f, poolLen);
  k_pool_feat<<<nblk(Nn * 64), TPB, 0, stream>>>(bufB, batch, Nn * 64, 6, sumsN);
  k_pool_cnt<<<nblk(Nn), TPB, 0, stream>>>(batch, Nn, cntN);
  if (Ee) {
    k_pool_feat_edge<<<nblk(Ee * 64), TPB, 0, stream>>>(bufD, srcN, batch, Ee * 64, 6, sumsE);
    k_pool_cnt_edge<<<nblk(Ee), TPB, 0, stream>>>(srcN, batch, Ee, cntE);
  }
  k_build_g<<<nblk((long long)Bb * 128), TPB, 0, stream>>>(sumsN, cntN, sumsE, cntE,
                                                           g, Bb * 128);

  // ---- final FC: out = g @ Wfc + bfc ----
  gemm(g, Wfc, tmp, Bb, DON, DON + DOE);                   // (64x128)@(128x64)
  k_bias_out<<<nblk((long long)Bb * DON), TPB, 0, stream>>>(tmp, bfc, (float*)d_out,
                                                            Bb * DON);
}